// AutoCorrelation_56470230007872
// MI455X (gfx1250) — compile-verified
//
#include <hip/hip_runtime.h>
#include <math.h>

// ---------------------------------------------------------------------------
// AutoCorrelation (Autoformer) for MI455X / gfx1250.
//
// corr[e,tau] = sum_t q[e,t] * k[e,(t-tau) mod 96]   (per b,n,h,e)
// computed via the DFT route so it becomes dense GEMMs with a SHARED basis:
//   Qs = Q * Wf ; Ks = K * Wf           (M=64, K=96, N=112: 49 cos | 49 sin)
//   P  = Qs (*) conj(Ks)                (pointwise complex)
//   C  = P * Wi                         (M=64, K=112, N=96; Wi = scaled Wf^T)
// All GEMMs run on V_WMMA_F32_16X16X4_F32 (fp32 keeps top-k lag selection
// faithful). V tile is staged LDS-direct with global_load_async_to_lds_b128
// overlapping the G2 WMMAs. Then top-4 + softmax + circular gather-aggregate.
// ---------------------------------------------------------------------------

typedef __attribute__((ext_vector_type(2))) float v2f;
typedef __attribute__((ext_vector_type(8))) float v8f;

#define LEN   96            // L (time / lag length)
#define CH    64            // E = D (channels per head)
#define NHEAD 8
#define NGRP  828           // B*N = 4*207
#define LQ    100           // padded stride for 96-wide rows (bank-safe)
#define LW    116           // padded stride for 112-wide rows (bank-safe)
#define SPEC  112           // 49 cos cols [0,49) + 49 sin cols [56,105), rest 0
#define SIN0  56
#define NF    49            // rfft bins for L=96

// LDS layout (dwords). sC aliases sQ (dead after G1); V tile ([t][e], stride
// 64) and top-k scratch alias sKs / sK (dead after pointwise / after G1).
#define OFF_Q   0                       // 64*100 = 6400
#define OFF_K   6400                    // 64*100 = 6400
#define OFF_WF  12800                   // 96*116 = 11136
#define OFF_QS  23936                   // 64*116 = 7424
#define OFF_KS  31360                   // 64*116 = 7424
#define SMEM_DW 38784                   // 155,136 bytes
#define OFF_C   0                       // aliases sQ  (64*100)
#define OFF_V2  OFF_KS                  // aliases sKs (96*64 = 6144 used)
#define OFF_TW  6400                    // aliases sK  (4*64 floats)
#define OFF_TD  6656                    // aliases sK  (4*64 ints)

__global__ __launch_bounds__(256)
void autocorr_kernel(const float* __restrict__ qg,
                     const float* __restrict__ kg,
                     const float* __restrict__ vg,
                     float* __restrict__ outV,
                     float* __restrict__ outC)
{
    __shared__ float smem[SMEM_DW];
    __shared__ float ctab[LEN];
    __shared__ float stab[LEN];

    const int tid  = threadIdx.x;
    const int lane = tid & 31;
    const int wave = tid >> 5;
    const int g    = blockIdx.x >> 3;   // (b*N + n)
    const int h    = blockIdx.x & 7;

    const size_t ioBase = (size_t)g * (LEN * NHEAD * CH) + (size_t)h * CH;

    float* sQ  = smem + OFF_Q;
    float* sK  = smem + OFF_K;
    float* sWf = smem + OFF_WF;
    float* sQs = smem + OFF_QS;
    float* sKs = smem + OFF_KS;
    float* sC  = smem + OFF_C;
    float* sV2 = smem + OFF_V2;         // [t][e], stride 64
    float* sTW = smem + OFF_TW;
    int*   sTD = (int*)(smem + OFF_TD);

    // ---- Phase 0a: twiddle tables + stage Q,K tiles transposed to [e][t] ---
    if (tid < LEN) {
        const float ang = (float)tid * (6.2831853071795864769f / (float)LEN);
        ctab[tid] = cosf(ang);
        stab[tid] = sinf(ang);
    }
    for (int i = tid; i < (LEN * CH) / 4; i += 256) {     // 6 iters, uniform
        const int t  = i >> 4;
        const int e4 = (i & 15) << 2;
        const size_t gi = ioBase + (size_t)t * (NHEAD * CH) + e4;
        __builtin_prefetch(&qg[gi + 16384], 0, 1);        // global_prefetch_b8
        __builtin_prefetch(&kg[gi + 16384], 0, 1);
        const float4 qv = *(const float4*)&qg[gi];
        const float4 kv = *(const float4*)&kg[gi];
        sQ[(e4 + 0) * LQ + t] = qv.x;  sK[(e4 + 0) * LQ + t] = kv.x;
        sQ[(e4 + 1) * LQ + t] = qv.y;  sK[(e4 + 1) * LQ + t] = kv.y;
        sQ[(e4 + 2) * LQ + t] = qv.z;  sK[(e4 + 2) * LQ + t] = kv.z;
        sQ[(e4 + 3) * LQ + t] = qv.w;  sK[(e4 + 3) * LQ + t] = kv.w;
    }
    __syncthreads();

    // ---- Phase 0b: build DFT basis from tables (exact angle reduction) -----
    for (int i = tid; i < LEN * SPEC; i += 256) {         // 42 iters, uniform
        const int t = i / SPEC, c = i - t * SPEC;
        float w = 0.0f;
        if (c < NF)                              w = ctab[(c * t) % LEN];
        else if (c >= SIN0 && c < SIN0 + NF)     w = stab[((c - SIN0) * t) % LEN];
        sWf[t * LW + c] = w;
    }
    __syncthreads();

    // ---- WMMA tile kernel: C[16x16] += A[16x4] * B[4x16] over Ksteps -------
    // A layout (ISA 7.12.2): lane<16 -> K={0,1}, lane>=16 -> K={2,3}, M=lane&15
    // B layout: VGPR0 -> rows K=0 (lanes 0-15) / K=2 (lanes 16-31), VGPR1 -> K=1/3
    // C layout: VGPR r -> M = r + (lane<16 ? 0 : 8), N = lane&15
    auto gemm_tile = [&](const float* A, int lda, int M0, int N0,
                         float* C, int ldc, int ksteps, bool inv) {
        const int half2 = (lane >> 4) << 1;               // 0 or 2
        const int mr    = lane & 15;
        v8f acc = {};
        for (int ks = 0; ks < ksteps; ++ks) {
            const int kk = ks * 4 + half2;
            v2f a, b;
            a.x = A[(M0 + mr) * lda + kk];
            a.y = A[(M0 + mr) * lda + kk + 1];
            if (!inv) {
                b.x = sWf[kk * LW + N0 + mr];
                b.y = sWf[(kk + 1) * LW + N0 + mr];
            } else {
                // Wi[kk][t] = alpha_f * Wf[t][kk]  (scaled transpose, in place)
                const int f0 = (kk < SIN0) ? kk : kk - SIN0;
                const int f1 = (kk + 1 < SIN0) ? kk + 1 : kk + 1 - SIN0;
                const float s0 = (f0 == 0 || f0 == 48) ? (1.0f / 96.0f) : (2.0f / 96.0f);
                const float s1 = (f1 == 0 || f1 == 48) ? (1.0f / 96.0f) : (2.0f / 96.0f);
                b.x = s0 * sWf[(N0 + mr) * LW + kk];
                b.y = s1 * sWf[(N0 + mr) * LW + kk + 1];
            }
            acc = __builtin_amdgcn_wmma_f32_16x16x4_f32(
                      false, a, false, b, (short)0, acc, false, false);
        }
        const int mb = M0 + ((lane >> 4) << 3);
        #pragma unroll
        for (int r = 0; r < 8; ++r)
            C[(mb + r) * ldc + (lane & 15)] = acc[r];
    };

    // ---- G1: Qs = Q*Wf, Ks = K*Wf.  2 inputs x 4 Mtiles x 7 Ntiles = 56 ----
    for (int tile = wave; tile < 56; tile += 8) {
        const int which = tile / 28;                      // 0 -> Q, 1 -> K
        const int t2 = tile - which * 28;
        const int Mt = t2 / 7, Nt = t2 - Mt * 7;
        gemm_tile(which ? sK : sQ, LQ, Mt * 16, Nt * 16,
                  which ? sKs : sQs, LW, LEN / 4, false);
    }
    __syncthreads();

    // ---- Pointwise: P = Qspec * conj(Kspec), written back into sQs --------
    // Each (e,f) pair (both its cos and sin columns) is owned by exactly one
    // thread, so in-place read-then-write needs no staging and no barrier.
    for (int i = tid; i < CH * NF; i += 256) {
        const int e = i / NF, f = i - e * NF;
        const float qc = sQs[e * LW + f],  qs = sQs[e * LW + SIN0 + f];
        const float kc = sKs[e * LW + f],  ks = sKs[e * LW + SIN0 + f];
        sQs[e * LW + f]        = qc * kc + qs * ks;       // Re
        sQs[e * LW + SIN0 + f] = qs * kc - qc * ks;       // -Im
    }
    __syncthreads();                                      // sKs now dead

    // ---- Async-stage V tile into sKs region as [t][e] (overlaps G2) -------
    // global_load_async_to_lds_b128: vdst = LDS byte offset, vaddr = 32-bit
    // byte offset, saddr = 64-bit base (GVS). Tracked by ASYNCcnt.
    {
        const unsigned long long vbase = (unsigned long long)(const void*)(vg + ioBase);
        const unsigned ldsV = (unsigned)(uintptr_t)(void*)sV2;  // low 32 = LDS offset
        for (int i = tid; i < (LEN * CH) / 4; i += 256) { // 6 iters, uniform
            const int t  = i >> 4;
            const int e4 = (i & 15) << 2;
            const unsigned voff = (unsigned)((t * (NHEAD * CH) + e4) * 4);
            const unsigned loff = ldsV + (unsigned)((t * CH + e4) * 4);
            asm volatile("global_load_async_to_lds_b128 %0, %1, %2"
                         :: "v"(loff), "v"(voff), "s"(vbase) : "memory");
        }
    }

    // ---- G2: corr = P * Wi.  4 Mtiles x 6 Ntiles = 24 tiles ---------------
    for (int tile = wave; tile < 24; tile += 8) {
        const int Mt = tile / 6, Nt = tile - Mt * 6;
        gemm_tile(sQs, LW, Mt * 16, Nt * 16, sC, LQ, SPEC / 4, true);
    }
    asm volatile("s_wait_asynccnt 0x0" ::: "memory");     // V tile landed
    __syncthreads();

    // ---- corr output (transposed [b,n,tau,h,e], coalesced) ----------------
    for (int i = tid; i < LEN * CH; i += 256) {
        const int tau = i >> 6, e = i & 63;
        outC[ioBase + (size_t)tau * (NHEAD * CH) + e] = sC[e * LQ + tau];
    }

    // ---- top-4 lags per channel + softmax (matches lax.top_k tie rule) ----
    if (tid < CH) {
        const int e = tid;
        float w0 = -1e30f, w1 = -1e30f, w2 = -1e30f, w3 = -1e30f;
        int   d0 = 0, d1 = 0, d2 = 0, d3 = 0;
        for (int t = 0; t < LEN; ++t) {
            const float x = sC[e * LQ + t];
            if (x > w3) {
                if (x > w0)      { w3=w2;d3=d2; w2=w1;d2=d1; w1=w0;d1=d0; w0=x;d0=t; }
                else if (x > w1) { w3=w2;d3=d2; w2=w1;d2=d1; w1=x;d1=t; }
                else if (x > w2) { w3=w2;d3=d2; w2=x;d2=t; }
                else             { w3=x;d3=t; }
            }
        }
        const float e1 = expf(w1 - w0);
        const float e2 = expf(w2 - w0);
        const float e3 = expf(w3 - w0);
        const float inv = 1.0f / (1.0f + e1 + e2 + e3);
        sTW[0 * 64 + e] = inv;       sTW[1 * 64 + e] = e1 * inv;
        sTW[2 * 64 + e] = e2 * inv;  sTW[3 * 64 + e] = e3 * inv;
        sTD[0 * 64 + e] = d0;  sTD[1 * 64 + e] = d1;
        sTD[2 * 64 + e] = d2;  sTD[3 * 64 + e] = d3;
    }
    __syncthreads();

    // ---- aggregate: V[t,d] = sum_i w_i[d] * v[(t+delay_i[d]) mod 96, d] ---
    // sV2 is [t][e] stride 64 -> bank == channel, conflict-free gather.
    for (int i = tid; i < LEN * CH; i += 256) {
        const int t = i >> 6, d = i & 63;
        float acc = 0.0f;
        #pragma unroll
        for (int j = 0; j < 4; ++j) {
            int idx = t + sTD[j * 64 + d];
            if (idx >= LEN) idx -= LEN;
            acc += sTW[j * 64 + d] * sV2[idx * CH + d];
        }
        outV[ioBase + (size_t)t * (NHEAD * CH) + d] = acc;
    }
}

extern "C" void kernel_launch(void* const* d_in, const int* in_sizes, int n_in,
                              void* d_out, int out_size, void* d_ws, size_t ws_size,
                              hipStream_t stream) {
    const float* q = (const float*)d_in[0];
    const float* k = (const float*)d_in[1];
    const float* v = (const float*)d_in[2];
    // d_in[3] = attn_mask (scalar 0), unused.

    float* outV = (float*)d_out;                               // [B,N,L,H,D]
    float* outC = outV + (size_t)NGRP * LEN * NHEAD * CH;      // [B,N,L,H,E]

    dim3 grid(NGRP * NHEAD);   // one workgroup per (b,n,h)
    dim3 block(256);           // 8 waves of 32
    autocorr_kernel<<<grid, block, 0, stream>>>(q, k, v, outV, outC);
}